// QuantizerDecoder_80539226734981
// MI455X (gfx1250) — compile-verified
//
#include <hip/hip_runtime.h>

typedef __attribute__((ext_vector_type(2))) float v2f;
typedef __attribute__((ext_vector_type(8))) float v8f;

// Problem constants (from reference)
#define MG   8      // groups
#define DD   128    // feature dim
#define KCB  4096   // codebook entries per group
#define HH   64
#define WW   64
#define NN   8
#define HW   (HH * WW)          // 4096, channel stride in output
#define TILES_TOTAL 2048        // (N*H*W)/16 token tiles (16 consecutive w each)
#define TILES_PER_BLOCK 8

// Block: 256 threads = 8 waves. blockIdx.y = m. Wave wv owns channels
// c0 = 16*wv .. c0+15 (8 waves x 16 = all 128 channels of group m).
//
// WMMA mapping (V_WMMA_F32_16X16X4_F32, D = A x B + C):
//   A (16x4)  = wq[m][c0+row][d..d+3]      rows = channels
//   B (4x16)  = gathered x^T               cols = 16 consecutive w tokens
//   D (16x16) = out[channel, token]        -> coalesced channel-major stores
//
// Per-lane fragment addressing (ISA 7.12.2, 32-bit 16x4 A layout):
//   lane L: row/col = L%16, K-pair = 2*(L/16); per k-step both A and B
//   fragments are a contiguous float2 at d = 4k + 2*(L/16).

__global__ __launch_bounds__(256)
void qdec_wmma_f32_kernel(const int*   __restrict__ codes,     // [N,H,W,M] int32
                          const float* __restrict__ codebook,  // [M,K,D]
                          const float* __restrict__ wq,        // [M,D,D]
                          const float* __restrict__ bq,        // [M,D]
                          float*       __restrict__ out)       // [N, M*D, H, W]
{
    const int lane = threadIdx.x & 31;
    const int wv   = threadIdx.x >> 5;     // 0..7
    const int m    = blockIdx.y;           // 0..7
    const int c0   = wv << 4;              // wave's channel base within group
    const int half = lane >> 4;            // 0 or 1 (selects K-pair / c+8 half)
    const int l16  = lane & 15;            // row (A) / token (B,D) index

    // ---- Preload A fragments: whole 16x128 wq tile for this wave ----
    // A[k] covers d = 4k..4k+3; this lane holds d = 4k + 2*half + {0,1}.
    const float* wq_row = wq + ((size_t)(m * DD + c0 + l16) * DD) + 2 * half;
    v2f A[32];
#pragma unroll
    for (int k = 0; k < 32; ++k) {
        A[k] = *(const v2f*)(wq_row + 4 * k);
    }

    // ---- Preload bias: D-VGPR v holds channel c0 + v + 8*half ----
    float bias[8];
#pragma unroll
    for (int v = 0; v < 8; ++v) {
        bias[v] = bq[m * DD + c0 + v + 8 * half];
    }

    const size_t cb_m = (size_t)m * KCB * DD;

    // ---- Token-tile loop ----
    for (int i = 0; i < TILES_PER_BLOCK; ++i) {
        const int tt  = blockIdx.x * TILES_PER_BLOCK + i;   // 0..2047
        const int w0  = (tt & 3) << 4;                      // w tile base
        const int row = tt >> 2;                            // n*64 + h
        const int h   = row & 63;
        const int n   = row >> 6;

        // This lane's token = l16; fetch its codebook index.
        const int code = codes[((size_t)row * WW + w0 + l16) * MG + m];
        // B fragment base: x^T, this lane holds d = 4k + 2*half + {0,1}.
        const float* xb = codebook + cb_m + (size_t)code * DD + 2 * half;

        v8f acc = {};
#pragma unroll
        for (int k = 0; k < 32; ++k) {
            v2f b = *(const v2f*)(xb + 4 * k);
            acc = __builtin_amdgcn_wmma_f32_16x16x4_f32(
                /*neg_a=*/false, A[k],
                /*neg_b=*/false, b,
                /*c_mod=*/(short)0, acc,
                /*reuse_a=*/false, /*reuse_b=*/false);
        }

        // ---- Store: out[n, m*128 + c, h, w0 + l16], c = c0 + v + 8*half ----
        // D-VGPR v, lanes 0-15: channel c0+v, token=lane; lanes 16-31: c0+v+8.
        const size_t base = ((size_t)(n * (MG * DD) + m * DD + c0 + 8 * half) * HW)
                          + (size_t)h * WW + w0 + l16;
#pragma unroll
        for (int v = 0; v < 8; ++v) {
            out[base + (size_t)v * HW] = acc[v] + bias[v];
        }
    }
}

extern "C" void kernel_launch(void* const* d_in, const int* in_sizes, int n_in,
                              void* d_out, int out_size, void* d_ws, size_t ws_size,
                              hipStream_t stream) {
    const int*   codes    = (const int*)  d_in[0];
    const float* codebook = (const float*)d_in[1];
    const float* wq       = (const float*)d_in[2];
    const float* bq       = (const float*)d_in[3];
    float*       out      = (float*)d_out;

    dim3 grid(TILES_TOTAL / TILES_PER_BLOCK, MG, 1);  // (256, 8)
    dim3 block(256, 1, 1);                            // 8 waves (wave32)
    qdec_wmma_f32_kernel<<<grid, block, 0, stream>>>(codes, codebook, wq, bq, out);
}